// MinesweeperGNN_29746943492174
// MI455X (gfx1250) — compile-verified
//
#include <hip/hip_runtime.h>
#include <hip/hip_bf16.h>

// CDNA5 / gfx1250: wave32, WMMA f32 16x16x4.
typedef float v2f __attribute__((ext_vector_type(2)));
typedef float v8f __attribute__((ext_vector_type(8)));

#define TB 256  // 8 waves per block (wave32)

__device__ __forceinline__ void atomic_add_f32(float* p, float v) {
#if defined(__AMDGCN__)
    unsafeAtomicAdd(p, v);   // lowers to global_atomic_add_f32 (no return)
#else
    atomicAdd(p, v);
#endif
}

// Force a wave-uniform predicate into an SGPR -> scalar branch, no saveexec.
__device__ __forceinline__ bool uniform_bool(bool c) {
#if defined(__AMDGCN__)
    return __builtin_amdgcn_readfirstlane((int)c) != 0;
#else
    return c;
#endif
}

// ---- init: deg = 1.0 (self loop), agg1 = agg2 = 0 (float4 stores) ---------
__global__ void gcn_init(float* __restrict__ deg, float4* __restrict__ agg1,
                         float4* __restrict__ agg2, long n8, int N) {
    long t = (long)blockIdx.x * blockDim.x + threadIdx.x;
    if (t < n8) {
        const float4 z = make_float4(0.f, 0.f, 0.f, 0.f);
        agg1[t] = z;
        agg2[t] = z;
        if (t < N) deg[t] = 1.0f;
    }
}

// ---- degree accumulation over dst -----------------------------------------
__global__ void gcn_degree(const int* __restrict__ dst, float* __restrict__ deg, int E) {
    long t = (long)blockIdx.x * blockDim.x + threadIdx.x;
    if (t < E) atomic_add_f32(&deg[dst[t]], 1.0f);
}

// ---- dinv = rsqrt(deg) (deg >= 1 always, keep reference guard) ------------
__global__ void gcn_rsqrt(float* __restrict__ deg, int N) {
    long t = (long)blockIdx.x * blockDim.x + threadIdx.x;
    if (t < N) {
        float d = deg[t];
        deg[t] = (d > 0.0f) ? rsqrtf(d) : 0.0f;
    }
}

// ---- layer-1 GEMM: h1[16-node tile, 32] = x[tile, 3(pad 4)] @ W1[3(pad 4), 32]
// One wave per 16-node tile, V_WMMA_F32_16X16X4_F32, K padded with zeros.
__global__ void gcn_gemm1_wmma(const float* __restrict__ x, const float* __restrict__ W1,
                               float* __restrict__ h1, int nTiles, int N) {
    const int wave = threadIdx.x >> 5;
    const int lane = threadIdx.x & 31;
    const int tile = blockIdx.x * (blockDim.x >> 5) + wave;
    if (tile >= nTiles) return;            // uniform per wave: EXEC stays full
    const int base = tile << 4;
    const int m    = lane & 15;
    const int hi   = lane >> 4;            // 0: K=0,1 half; 1: K=2,3 half
    const bool full = uniform_bool(base + 16 <= N);   // scalar branch below

    int node = base + m;
    if (!full && node >= N) node = N - 1;  // clamp tail reads only
    const float* xr = x + (long)node * 3;
    const float x0 = xr[0], x1 = xr[1], x2 = xr[2];   // one global_load_b96
    v2f a;
    a.x = hi ? x2   : x0;
    a.y = hi ? 0.0f : x1;

    float* outb = h1 + (long)base * 32 + (long)(hi << 3) * 32;   // row r / r+8

#pragma unroll
    for (int n = 0; n < 2; ++n) {          // output column groups 0-15 / 16-31
        const int col = (n << 4) + m;
        const float w0 = W1[col], w1 = W1[32 + col], w2 = W1[64 + col];
        v2f b;
        b.x = hi ? w2   : w0;
        b.y = hi ? 0.0f : w1;
        v8f c = {};
        c = __builtin_amdgcn_wmma_f32_16x16x4_f32(false, a, false, b,
                                                  (short)0, c, false, false);
        float* o = outb + col;
        if (full) {
#pragma unroll
            for (int r = 0; r < 8; ++r) o[r * 32] = c[r];    // imm-offset stores
        } else {
#pragma unroll
            for (int r = 0; r < 8; ++r)
                if (base + (hi << 3) + r < N) o[r * 32] = c[r];
        }
    }
}

// ---- layer-2 GEMM: hout[tile,32] = hin[tile,32] @ W2[32,32]  (16 WMMAs/wave)
__global__ void gcn_gemm2_wmma(const float* __restrict__ hin, const float* __restrict__ W2,
                               float* __restrict__ hout, int nTiles, int N) {
    const int wave = threadIdx.x >> 5;
    const int lane = threadIdx.x & 31;
    const int tile = blockIdx.x * (blockDim.x >> 5) + wave;
    if (tile >= nTiles) return;
    const int base = tile << 4;
    const int m    = lane & 15;
    const int hi   = lane >> 4;
    const int koff = hi << 1;              // 0 or 2
    const bool full = uniform_bool(base + 16 <= N);

    int node = base + m;
    if (!full && node >= N) node = N - 1;
    const float* hr = hin + (long)node * 32 + koff;   // 8B-aligned (koff even)

    v2f a[8];
#pragma unroll
    for (int kk = 0; kk < 8; ++kk) {       // K chunks of 4; global_load_b64 each
        const float2 t = *(const float2*)(hr + 4 * kk);
        a[kk].x = t.x; a[kk].y = t.y;
    }

    float* outb = hout + (long)base * 32 + (long)(hi << 3) * 32;

#pragma unroll
    for (int n = 0; n < 2; ++n) {
        const int col = (n << 4) + m;
        const float* wp = W2 + (long)koff * 32 + col;  // fold koff into base ptr
        v8f c = {};
#pragma unroll
        for (int kk = 0; kk < 8; ++kk) {
            v2f b;
            b.x = wp[kk * 128];            // row 4kk+koff
            b.y = wp[kk * 128 + 32];       // row 4kk+koff+1
            c = __builtin_amdgcn_wmma_f32_16x16x4_f32(false, a[kk], false, b,
                                                      (short)0, c, false, false);
        }
        float* o = outb + col;
        if (full) {
#pragma unroll
            for (int r = 0; r < 8; ++r) o[r * 32] = c[r];
        } else {
#pragma unroll
            for (int r = 0; r < 8; ++r)
                if (base + (hi << 3) + r < N) o[r * 32] = c[r];
        }
    }
}

// ---- edge scatter: agg[dst] += h[src] * dinv[src]*dinv[dst]; 4 threads/edge
__global__ void gcn_scatter(const int* __restrict__ src, const int* __restrict__ dst,
                            const float* __restrict__ dinv, const float* __restrict__ h,
                            float* __restrict__ agg, int E) {
    long t = (long)blockIdx.x * blockDim.x + threadIdx.x;
    long e = t >> 2;
    if (e >= E) return;
    const int part = (int)(t & 3);                 // 8 features per thread
    const int s = src[e], d = dst[e];
    const float w = dinv[s] * dinv[d];
    const float4* hs = (const float4*)(h + (long)s * 32 + part * 8);
    const float4 v0 = hs[0], v1 = hs[1];
    float* o = agg + (long)d * 32 + part * 8;
    atomic_add_f32(o + 0, v0.x * w);
    atomic_add_f32(o + 1, v0.y * w);
    atomic_add_f32(o + 2, v0.z * w);
    atomic_add_f32(o + 3, v0.w * w);
    atomic_add_f32(o + 4, v1.x * w);
    atomic_add_f32(o + 5, v1.y * w);
    atomic_add_f32(o + 6, v1.z * w);
    atomic_add_f32(o + 7, v1.w * w);
}

// ---- fold self-loop + bias + relu, in place over h (float4) ---------------
// h[i,f] = relu(agg[i,f] + dinv[i]^2 * h[i,f] + b[f])
__global__ void gcn_relu_self_bias(const float4* __restrict__ agg, const float* __restrict__ dinv,
                                   const float* __restrict__ b, float4* __restrict__ h, long n8) {
    long t = (long)blockIdx.x * blockDim.x + threadIdx.x;
    if (t >= n8) return;
    const long i = t >> 3;                 // node (8 float4 groups per node)
    const int fg = (int)(t & 7) << 2;      // feature group start
    const float c  = dinv[i];
    const float cc = c * c;
    const float4 A = agg[t];
    const float4 H = h[t];
    const float4 B = *(const float4*)(b + fg);
    float4 r;
    r.x = fmaxf(A.x + cc * H.x + B.x, 0.0f);
    r.y = fmaxf(A.y + cc * H.y + B.y, 0.0f);
    r.z = fmaxf(A.z + cc * H.z + B.z, 0.0f);
    r.w = fmaxf(A.w + cc * H.w + B.w, 0.0f);
    h[t] = r;
}

// ---- output head: out[i,:2] = h2[i,:32] @ W3[32,2] + b3 -------------------
__global__ void gcn_out(const float* __restrict__ h2, const float* __restrict__ W3,
                        const float* __restrict__ b3, float* __restrict__ out, int N) {
    long i = (long)blockIdx.x * blockDim.x + threadIdx.x;
    if (i >= N) return;
    const float* hr = h2 + i * 32;
    float a0 = b3[0], a1 = b3[1];
#pragma unroll
    for (int k = 0; k < 32; ++k) {
        const float v = hr[k];
        a0 += v * W3[k * 2 + 0];
        a1 += v * W3[k * 2 + 1];
    }
    out[i * 2 + 0] = a0;
    out[i * 2 + 1] = a1;
}

extern "C" void kernel_launch(void* const* d_in, const int* in_sizes, int n_in,
                              void* d_out, int out_size, void* d_ws, size_t ws_size,
                              hipStream_t stream) {
    const float* x  = (const float*)d_in[0];
    const int*   ei = (const int*)  d_in[1];
    const float* W1 = (const float*)d_in[2];
    const float* b1 = (const float*)d_in[3];
    const float* W2 = (const float*)d_in[4];
    const float* b2 = (const float*)d_in[5];
    const float* W3 = (const float*)d_in[6];
    const float* b3 = (const float*)d_in[7];
    float* out = (float*)d_out;

    const int N = in_sizes[0] / 3;
    const int E = in_sizes[1] / 2;
    const int* src = ei;         // edge_index[0]
    const int* dst = ei + E;     // edge_index[1]

    // Workspace carve-up: dinv[N] | B1 (h1/h1r) | B2 (agg1/h2pre/h2) | B3 (agg2)
    char* ws = (char*)d_ws;
    const size_t featBytes = (size_t)N * 32 * sizeof(float);
    size_t off = (((size_t)N * sizeof(float)) + 255) & ~(size_t)255;
    float* dinv = (float*)ws;
    float* B1 = (float*)(ws + off);
    float* B2 = (float*)(ws + off + featBytes);
    float* B3 = (float*)(ws + off + 2 * featBytes);

    const long n8  = (long)N * 8;          // float4 groups in a feature buffer
    const int g_n8    = (int)((n8 + TB - 1) / TB);
    const int g_E     = (E + TB - 1) / TB;
    const int g_E4    = (int)(((long)E * 4 + TB - 1) / TB);
    const int nTiles  = (N + 15) / 16;
    const int g_tiles = (nTiles + (TB / 32) - 1) / (TB / 32);
    const int g_N     = (N + TB - 1) / TB;

    // deg (with self loop) -> dinv
    gcn_init  <<<g_n8, TB, 0, stream>>>(dinv, (float4*)B2, (float4*)B3, n8, N);
    gcn_degree<<<g_E,  TB, 0, stream>>>(dst, dinv, E);
    gcn_rsqrt <<<g_N,  TB, 0, stream>>>(dinv, N);

    // layer 1: h1 = x@W1 ; agg1 = scatter ; h1r = relu(agg1 + selfloop + b1)
    gcn_gemm1_wmma    <<<g_tiles, TB, 0, stream>>>(x, W1, B1, nTiles, N);
    gcn_scatter       <<<g_E4,    TB, 0, stream>>>(src, dst, dinv, B1, B2, E);
    gcn_relu_self_bias<<<g_n8,    TB, 0, stream>>>((const float4*)B2, dinv, b1, (float4*)B1, n8);

    // layer 2: h2pre = h1r@W2 ; agg2 = scatter ; h2 = relu(agg2 + selfloop + b2)
    gcn_gemm2_wmma    <<<g_tiles, TB, 0, stream>>>(B1, W2, B2, nTiles, N);
    gcn_scatter       <<<g_E4,    TB, 0, stream>>>(src, dst, dinv, B2, B3, E);
    gcn_relu_self_bias<<<g_n8,    TB, 0, stream>>>((const float4*)B3, dinv, b2, (float4*)B2, n8);

    // head
    gcn_out<<<g_N, TB, 0, stream>>>(B2, W3, b3, out, N);
}